// PWLUBase_3659312136864
// MI455X (gfx1250) — compile-verified
//
#include <hip/hip_runtime.h>

// PWLU forward, channelwise piecewise-linear activation.
// x: [B=32, C=256, H=56, W=56] f32; points: [C, 13] f32; bounds: [C, 2] f32 -> out f32
//
// Memory-bound: 25.7M elems * 8 B = 205 MB -> ~8.8 us floor @ 23.3 TB/s.
//  - one block per (b,c) plane: channel table/bounds are block-uniform (SGPR loads)
//  - 13-knot table repacked in LDS as 12 float2 segments {p[i], p[i+1]}:
//    the per-lane gather is one aligned ds_load_b64
//  - H*W/4 = 784 = 3*256 + 16: fully unrolled, all B128 NT loads issued before
//    the barrier so they overlap the LDS table setup
//  - nontemporal B128 loads/stores keep the 2x103 MB streams from thrashing L2

#define N_REGIONS 12
#define HW4       784      // (56*56)/4
#define BLOCK     256
#define TAIL      (HW4 - 3 * BLOCK)   // 16

typedef float v4f __attribute__((ext_vector_type(4)));

__device__ __forceinline__ v4f pwlu4(v4f v, float lo, float inv, const float2* seg) {
    v4f r;
#pragma unroll
    for (int k = 0; k < 4; ++k) {
        float xn = (v[k] - lo) * inv;                                  // normalize
        float t  = xn * (float)N_REGIONS;
        float rf = floorf(fminf(fmaxf(xn, 0.0f), 0.999f) * (float)N_REGIONS);
        float d  = t - rf;                                             // extrapolates outside bounds
        float2 pr = seg[(int)rf];                                      // ds_load_b64, 8B aligned
        r[k] = pr.x * (1.0f - d) + pr.y * d;                           // reference lerp form
    }
    return r;
}

__global__ __launch_bounds__(BLOCK, 1) void pwlu_fwd_kernel(
    const float* __restrict__ x,
    const float* __restrict__ points,   // [C, N_REGIONS+1]
    const float* __restrict__ bounds,   // [C, 2]
    float* __restrict__ out,
    int cmask)                          // C-1 (C power of two)
{
    __shared__ float2 seg[N_REGIONS];   // seg[i] = {points[c][i], points[c][i+1]}

    const int plane = blockIdx.x;       // plane = b*C + c
    const int c     = plane & cmask;
    const int tid   = threadIdx.x;

    if (tid < N_REGIONS) {
        const float* p = points + (size_t)c * (N_REGIONS + 1) + tid;
        seg[tid] = make_float2(p[0], p[1]);
    }
    // Block-uniform scalars -> s_load path.
    const float lo  = bounds[2 * c + 0];
    const float hi  = bounds[2 * c + 1];
    const float inv = 1.0f / (hi - lo);

    const v4f* __restrict__ xin  = (const v4f*)(x   + (size_t)plane * (size_t)(HW4 * 4));
    v4f*       __restrict__ yout = (v4f*)      (out + (size_t)plane * (size_t)(HW4 * 4));

    // Issue all bulk B128 NT loads before the barrier (overlap with table setup).
    v4f a0 = __builtin_nontemporal_load(xin + tid);
    v4f a1 = __builtin_nontemporal_load(xin + tid + BLOCK);
    v4f a2 = __builtin_nontemporal_load(xin + tid + 2 * BLOCK);
    v4f a3 = {};
    const bool tail = tid < TAIL;
    if (tail) a3 = __builtin_nontemporal_load(xin + tid + 3 * BLOCK);

    __syncthreads();

    __builtin_nontemporal_store(pwlu4(a0, lo, inv, seg), yout + tid);
    __builtin_nontemporal_store(pwlu4(a1, lo, inv, seg), yout + tid + BLOCK);
    __builtin_nontemporal_store(pwlu4(a2, lo, inv, seg), yout + tid + 2 * BLOCK);
    if (tail)
        __builtin_nontemporal_store(pwlu4(a3, lo, inv, seg), yout + tid + 3 * BLOCK);
}

extern "C" void kernel_launch(void* const* d_in, const int* in_sizes, int n_in,
                              void* d_out, int out_size, void* d_ws, size_t ws_size,
                              hipStream_t stream) {
    (void)n_in; (void)d_ws; (void)ws_size; (void)out_size;

    const float* x      = (const float*)d_in[0];
    const float* points = (const float*)d_in[1];
    const float* bounds = (const float*)d_in[2];
    float*       out    = (float*)d_out;

    const int C      = in_sizes[1] / (N_REGIONS + 1);  // 256 (power of two)
    const int planes = in_sizes[0] / (HW4 * 4);        // B*C = 8192

    pwlu_fwd_kernel<<<planes, BLOCK, 0, stream>>>(x, points, bounds, out, C - 1);
}